// GetNodeK_7335804141780
// MI455X (gfx1250) — compile-verified
//
#include <hip/hip_runtime.h>
#include <cstdint>
#include <cstddef>

// ---------------------------------------------------------------------------
// GetNodeK gather: out[b,i,n,m,:] = emb[b, nbr[b, nbr[b,i,n], m], :]
// B=8, At=256, Nbr=24, F=128. Zero FLOPs, 604 MB out -> store-BW bound
// (604 MB / 23.3 TB/s ~= 26 us floor on MI455X).
// Strategy: per-workgroup LDS residency of emb[b] (128KB) + nbr[b] (24KB),
// staged with gfx1250 async-to-LDS copies (ASYNCcnt), output rows emitted
// straight from LDS with async-from-LDS b128 stores (no VGPR round trip),
// non-temporal so the 604 MB write stream doesn't evict the hot input
// tables from the 192 MB L2.
// ---------------------------------------------------------------------------

namespace {
constexpr int kB   = 8;
constexpr int kAt  = 256;
constexpr int kNbr = 24;
constexpr int kF   = 128;

constexpr int kIPG     = 4;             // atoms (i) handled per workgroup
constexpr int kChunks  = kAt / kIPG;    // 64 chunks per batch -> 512 WGs
constexpr int kThreads = 256;           // 8 wave32 waves per workgroup
constexpr int kWaves   = kThreads / 32;

constexpr int    kEmbFloats = kAt * kF;    // 32768 floats = 128 KB
constexpr int    kIdxInts   = kAt * kNbr;  // 6144 ints   =  24 KB
constexpr size_t kLdsBytes  = size_t(kEmbFloats) * 4 + size_t(kIdxInts) * 4; // 155,648 B

// gfx12-family CPol: TH[2:0] | SCOPE[4:3].  TH=1 -> non-temporal.
constexpr int kCpolRT = 0;   // regular temporal (loads: keep emb/idx L2-hot)
constexpr int kCpolNT = 1;   // non-temporal   (stores: 604 MB write-once stream)
}  // namespace

typedef float float4v __attribute__((vector_size(16)));
// Builtin parameter type per hipcc diagnostic: vector_size(16) int, AS1/AS3.
typedef int v4i __attribute__((vector_size(16)));
typedef __attribute__((address_space(1))) v4i v4i_g;   // global
typedef __attribute__((address_space(3))) v4i v4i_s;   // LDS

#if defined(__gfx1250__) && __has_builtin(__builtin_amdgcn_global_load_async_to_lds_b128) && \
    __has_builtin(__builtin_amdgcn_s_wait_asynccnt)
#define GNK_ASYNC_LOAD 1
#else
#define GNK_ASYNC_LOAD 0
#endif

#if defined(__gfx1250__) && __has_builtin(__builtin_amdgcn_global_store_async_from_lds_b128) && \
    __has_builtin(__builtin_amdgcn_s_wait_asynccnt)
#define GNK_ASYNC_STORE 1
#else
#define GNK_ASYNC_STORE 0
#endif

__global__ __launch_bounds__(kThreads)
void GetNodeK_gather_kernel(const float* __restrict__ emb,
                            const int* __restrict__ nbr,
                            float* __restrict__ out) {
  extern __shared__ char smem[];
  float* s_emb = reinterpret_cast<float*>(smem);
  int*   s_idx = reinterpret_cast<int*>(smem + size_t(kEmbFloats) * 4);

  const int blk = blockIdx.x;
  const int b   = blk / kChunks;
  const int i0  = (blk % kChunks) * kIPG;
  const int tid = threadIdx.x;

  const float* embB = emb + size_t(b) * kEmbFloats;
  const int*   nbrB = nbr + size_t(b) * kIdxInts;

  // ---- Phase 1: stage emb[b] (128 KB) and nbr[b] (24 KB) into LDS ----------
#if GNK_ASYNC_LOAD
  for (int v = tid; v < kEmbFloats / 4; v += kThreads) {
    __builtin_amdgcn_global_load_async_to_lds_b128(
        (v4i_g*)(embB + v * 4),
        (v4i_s*)(s_emb + v * 4),
        /*offset=*/0, kCpolRT);
  }
  for (int v = tid; v < kIdxInts / 4; v += kThreads) {
    __builtin_amdgcn_global_load_async_to_lds_b128(
        (v4i_g*)(nbrB + v * 4),
        (v4i_s*)(s_idx + v * 4),
        /*offset=*/0, kCpolRT);
  }
  __builtin_amdgcn_s_wait_asynccnt(0);   // my async copies landed in LDS
#else
  for (int v = tid; v < kEmbFloats / 4; v += kThreads) {
    reinterpret_cast<float4v*>(s_emb)[v] =
        reinterpret_cast<const float4v*>(embB)[v];
  }
  for (int v = tid; v < kIdxInts / 4; v += kThreads) {
    reinterpret_cast<float4v*>(s_idx)[v] =
        reinterpret_cast<const float4v*>(nbrB)[v];
  }
#endif
  __syncthreads();                       // everyone's copies visible

  // ---- Phase 2: chained index resolve + row stream-out ---------------------
  // One wave owns one 512 B output row: lane l moves bytes [16l, 16l+16).
  const int lane    = tid & 31;
  const int wave    = tid >> 5;
  const int laneOff = lane * 4;          // float offset within a row

  for (int pair = wave; pair < kIPG * kNbr; pair += kWaves) {
    const int di = pair / kNbr;
    const int n  = pair % kNbr;
    const int i  = i0 + di;
    const int j  = s_idx[i * kNbr + n];            // nbr[b, i, n]
    const int* jrow = s_idx + j * kNbr;
    float* dstRow =
        out + ((((size_t)b * kAt + i) * kNbr + n) * kNbr) * kF + laneOff;

#pragma unroll
    for (int m = 0; m < kNbr; ++m) {
      const int k = jrow[m];                        // nbr[b, j, m]
#if GNK_ASYNC_STORE
      __builtin_amdgcn_global_store_async_from_lds_b128(
          (v4i_g*)(dstRow + (size_t)m * kF),
          (v4i_s*)(s_emb + k * kF + laneOff),
          /*offset=*/0, kCpolNT);
#else
      float4v t = *reinterpret_cast<const float4v*>(s_emb + k * kF + laneOff);
      *reinterpret_cast<float4v*>(dstRow + (size_t)m * kF) = t;
#endif
    }
  }

#if GNK_ASYNC_STORE
  __builtin_amdgcn_s_wait_asynccnt(0);   // drain LDS->global stores
#endif
}

extern "C" void kernel_launch(void* const* d_in, const int* in_sizes, int n_in,
                              void* d_out, int out_size, void* d_ws, size_t ws_size,
                              hipStream_t stream) {
  (void)in_sizes; (void)n_in; (void)out_size; (void)d_ws; (void)ws_size;

  const float* emb = (const float*)d_in[0];   // (B, At, F) f32
  const int*   nbr = (const int*)d_in[1];     // (B, At, Nbr) int
  float*       out = (float*)d_out;           // (B, At, Nbr, Nbr, F) f32

  // 152 KB dynamic LDS > default cap: raise the limit (idempotent, capture-safe).
  (void)hipFuncSetAttribute(reinterpret_cast<const void*>(GetNodeK_gather_kernel),
                            hipFuncAttributeMaxDynamicSharedMemorySize,
                            (int)kLdsBytes);

  dim3 grid(kB * kChunks);   // 512 workgroups
  GetNodeK_gather_kernel<<<grid, kThreads, kLdsBytes, stream>>>(emb, nbr, out);
}